// RNN_18056042512798
// MI455X (gfx1250) — compile-verified
//
#include <hip/hip_runtime.h>
#include <hip/hip_bf16.h>

// Problem dimensions (fixed by the reference).
constexpr int U_ROWS  = 2048;   // UNIT
constexpr int B_COLS  = 4096;   // B
constexpr int T_STEPS = 1024;   // T

// Blocking: 8 wave32's per block share 32 b-columns; each thread owns NU u-rows.
constexpr int LANES       = 32;                 // b-columns per block
constexpr int WAVES       = 8;                  // waves per block (256 threads)
constexpr int NU          = 8;                  // u-rows per thread
constexpr int U_PER_BLOCK = WAVES * NU;         // 64
constexpr int T_TILE      = 128;                // timesteps staged per LDS tile
constexpr int TILE_ELEMS  = T_TILE * LANES;     // 4096 floats = 16 KB per buffer
constexpr int N_TILES     = T_STEPS / T_TILE;   // 8

// ---- CDNA5 feature probes (compile-safe fallbacks) -------------------------
#if defined(__HIP_DEVICE_COMPILE__) && __has_builtin(__builtin_amdgcn_global_load_async_to_lds_b32)
#define HAVE_ASYNC_LDS 1
#else
#define HAVE_ASYNC_LDS 0
#endif

#if defined(__HIP_DEVICE_COMPILE__) && __has_builtin(__builtin_amdgcn_global_load_async_to_lds_b128)
#define HAVE_ASYNC_B128 1
#else
#define HAVE_ASYNC_B128 0
#endif

typedef int v4i __attribute__((ext_vector_type(4)));
typedef __attribute__((address_space(1))) int* gptr_i32;
typedef __attribute__((address_space(3))) int* lptr_i32;
typedef __attribute__((address_space(1))) v4i* gptr_v4;
typedef __attribute__((address_space(3))) v4i* lptr_v4;

__device__ __forceinline__ float htanh(float x) {
#if defined(__HIP_DEVICE_COMPILE__) && __has_builtin(__builtin_amdgcn_tanhf)
    return __builtin_amdgcn_tanhf(x);      // V_TANH_F32, TRANS pipe
#else
    return tanhf(x);                        // ocml software fallback
#endif
}

__device__ __forceinline__ void stage_b32(const float* gsrc, float* lds_dst) {
#if HAVE_ASYNC_LDS
    __builtin_amdgcn_global_load_async_to_lds_b32(
        (gptr_i32)(void*)gsrc, (lptr_i32)(void*)lds_dst, 0, 0);
#else
    *lds_dst = *gsrc;
#endif
}

__device__ __forceinline__ void stage_b128(const float* gsrc, float* lds_dst) {
#if HAVE_ASYNC_B128
    __builtin_amdgcn_global_load_async_to_lds_b128(
        (gptr_v4)(void*)gsrc, (lptr_v4)(void*)lds_dst, 0, 0);
#else
    stage_b32(gsrc + 0, lds_dst + 0);
    stage_b32(gsrc + 1, lds_dst + 1);
    stage_b32(gsrc + 2, lds_dst + 2);
    stage_b32(gsrc + 3, lds_dst + 3);
#endif
}

__device__ __forceinline__ void wait_stage() {
#if HAVE_ASYNC_LDS
#if __has_builtin(__builtin_amdgcn_s_wait_asynccnt)
    __builtin_amdgcn_s_wait_asynccnt(0);
#else
    asm volatile("s_wait_asynccnt 0" ::: "memory");
#endif
#endif
}

// Stage one x tile [t0, t0+T_TILE) x [b0, b0+LANES) into LDS buffer `buf`.
// XT=true : x is (T,B) transposed -> 16B-contiguous, use b128 async DMA.
// XT=false: x is (B,T) original   -> strided, scalar b32 async DMA.
template <bool XT>
__device__ __forceinline__ void stage_tile(const float* __restrict__ x, float* buf,
                                           int t0, int b0, int tid) {
    if (XT) {
#pragma unroll
        for (int k = 0; k < TILE_ELEMS / 4 / 256; ++k) {   // 4 float4 per thread
            const int e4 = k * 256 + tid;
            const int tl = e4 >> 3;                        // 8 float4 per 32-col row
            const int c  = (e4 & 7) * 4;
            stage_b128(x + (size_t)(t0 + tl) * B_COLS + b0 + c, &buf[e4 * 4]);
        }
    } else {
#pragma unroll
        for (int k = 0; k < TILE_ELEMS / 256; ++k) {
            const int e  = k * 256 + tid;
            const int tl = e >> 5;
            const int bl = e & 31;
            stage_b32(x + (size_t)(b0 + bl) * T_STEPS + t0 + tl, &buf[e]);
        }
    }
}

// ---- Transpose inputs (B,T) -> xT (T,B) so staging loads are coalesced -----
__global__ __launch_bounds__(256) void xpose_kernel(const float* __restrict__ in,
                                                    float* __restrict__ out) {
    __shared__ float tile[32][33];
    const int tx = threadIdx.x;            // 0..31
    const int ty = threadIdx.y;            // 0..7
    const int t_base = blockIdx.x * 32;
    const int b_base = blockIdx.y * 32;
#pragma unroll
    for (int i = 0; i < 4; ++i) {
        int bb = ty + i * 8;
        tile[bb][tx] = in[(size_t)(b_base + bb) * T_STEPS + t_base + tx];
    }
    __syncthreads();
#pragma unroll
    for (int i = 0; i < 4; ++i) {
        int tt = ty + i * 8;
        out[(size_t)(t_base + tt) * B_COLS + b_base + tx] = tile[tx][tt];
    }
}

// ---- Main elementwise-recurrence kernel ------------------------------------
template <bool XT>
__global__ __launch_bounds__(256) void rnn_kernel(const float* __restrict__ x,
                                                  const float* __restrict__ Wx,
                                                  const float* __restrict__ Wh,
                                                  const float* __restrict__ bias,
                                                  float* __restrict__ out) {
    __shared__ float xs[2][TILE_ELEMS];    // double buffer, 32 KB total

    const int tid  = threadIdx.x;
    const int lane = tid & (LANES - 1);
    const int wave = tid >> 5;
    const int b0   = blockIdx.x * LANES;
    const int b    = b0 + lane;
    const int u0   = blockIdx.y * U_PER_BLOCK + wave * NU;

    float wx[NU], wh[NU], bi[NU], h[NU];
#pragma unroll
    for (int k = 0; k < NU; ++k) {
        const size_t idx = (size_t)(u0 + k) * B_COLS + b;   // coalesced per wave
        wx[k] = Wx[idx];
        wh[k] = Wh[idx];
        bi[k] = bias[idx];
        h[k]  = 0.0f;
    }

    // Prologue: stage tile 0.
    stage_tile<XT>(x, xs[0], 0, b0, tid);
    wait_stage();
    __syncthreads();

    for (int i = 0; i < N_TILES; ++i) {
        // Prefetch next tile into the other buffer; overlaps with compute below.
        if (i + 1 < N_TILES)
            stage_tile<XT>(x, xs[(i + 1) & 1], (i + 1) * T_TILE, b0, tid);

        const float* buf = xs[i & 1];
#pragma unroll 4
        for (int tl = 0; tl < T_TILE; ++tl) {
            const float xv = buf[tl * LANES + lane];   // conflict-free broadcast
#pragma unroll
            for (int j = 0; j < NU; ++j)
                h[j] = htanh(fmaf(wh[j], h[j], fmaf(wx[j], xv, bi[j])));
        }

        wait_stage();       // prefetch finished long ago, behind the tanh chain
        __syncthreads();
    }

    // Extra step reusing x at t = 0 (reference's final call()).
    {
        const float xv = XT ? x[b] : x[(size_t)b * T_STEPS];
#pragma unroll
        for (int j = 0; j < NU; ++j)
            h[j] = htanh(fmaf(wh[j], h[j], fmaf(wx[j], xv, bi[j])));
    }

#pragma unroll
    for (int k = 0; k < NU; ++k)
        out[(size_t)(u0 + k) * B_COLS + b] = h[k];
}

// ---- Launch ----------------------------------------------------------------
extern "C" void kernel_launch(void* const* d_in, const int* in_sizes, int n_in,
                              void* d_out, int out_size, void* d_ws, size_t ws_size,
                              hipStream_t stream) {
    const float* inputs = (const float*)d_in[0];   // (B, T)
    const float* Wx     = (const float*)d_in[1];   // (U, B)
    const float* Wh     = (const float*)d_in[2];   // (U, B)
    const float* bias   = (const float*)d_in[3];   // (U, B)
    float* out          = (float*)d_out;           // (U, B)

    const size_t xt_bytes = (size_t)T_STEPS * B_COLS * sizeof(float);
    const dim3 block(256);
    const dim3 grid(B_COLS / LANES, U_ROWS / U_PER_BLOCK);   // (128, 32)

    if (d_ws && ws_size >= xt_bytes) {
        float* xT = (float*)d_ws;                  // (T, B)
        xpose_kernel<<<dim3(T_STEPS / 32, B_COLS / 32), dim3(32, 8), 0, stream>>>(inputs, xT);
        rnn_kernel<true><<<grid, block, 0, stream>>>(xT, Wx, Wh, bias, out);
    } else {
        rnn_kernel<false><<<grid, block, 0, stream>>>(inputs, Wx, Wh, bias, out);
    }
}